// Texture2D_1047972021062
// MI455X (gfx1250) — compile-verified
//
#include <hip/hip_runtime.h>
#include <hip/hip_bf16.h>

// ---------------------------------------------------------------------------
// Mip-mapped texture sampling (nvdiffrast dr.texture, linear-mipmap-linear,
// boundary 'wrap') for MI455X / gfx1250.
//
// Phase 1: build 11 mip levels (1024^2 .. 1^2, 3ch fp32) in d_ws via 2x2 avg
//          pooling. Each workgroup stages its 2-row source tile into LDS with
//          one TDM tensor_load_to_lds (CDNA5 DMA engine), waits TENSORcnt,
//          barriers, pools from LDS. Fallback: direct b96 gathers.
// Phase 2: per-pixel LOD from uv_da, bilinear-wrap sample two mip levels
//          (power-of-two dims -> & (D-1) wrap), trilinear lerp, b96 store.
//
// ws usage: 4,194,303 floats (~16.8 MB) for the pyramid.
// ---------------------------------------------------------------------------

typedef unsigned int v4u __attribute__((ext_vector_type(4)));
typedef int          v4i __attribute__((ext_vector_type(4)));
typedef int          v8i __attribute__((ext_vector_type(8)));

#define TEX_DIM   2048
#define N_LEVELS  12          // levels 0..11 (2048 -> 1)
#define OUT_PIX   (1024 * 1024)
#define POOL_BW   128         // output texels per pool workgroup (4 waves)

struct F3 { float x, y, z; };
struct MipOff { unsigned off[N_LEVELS]; };   // float offsets into ws, level>=1

// ---------------------------------------------------------------------------
// Phase 1: 2x2 average pool  (src: Ws x Ws x 3)  ->  (dst: Wd x Wd x 3)
// ---------------------------------------------------------------------------
__global__ void mip_pool_kernel(const float* __restrict__ src,
                                float* __restrict__ dst,
                                int Ws, int Wd) {
  const int tid  = threadIdx.x;
  const int x0   = blockIdx.x * POOL_BW;   // first output texel of this tile
  const int yo   = blockIdx.y;             // output row
  const int nout = min(POOL_BW, Wd - x0);  // output texels in this tile

#if defined(__gfx1250__) && __has_builtin(__builtin_amdgcn_tensor_load_to_lds)
  // --- CDNA5 path: TDM DMA of the 2-row source tile into LDS -------------
  __shared__ float tile[2 * POOL_BW * 2 * 3];  // packed [2][tw] floats, 6 KB
  const int tw = nout * 6;                     // floats per source row in tile

  if (tid < 32) {  // wave 0 issues the tensor DMA (wave-level op)
    const float* gsrc =
        src + ((size_t)yo * 2 * Ws + (size_t)x0 * 2) * 3;
    unsigned long long ga  = (unsigned long long)(const void*)gsrc;
    unsigned           lds = (unsigned)(unsigned long long)(void*)tile;

    // D# group 0: count=1 | lds_addr[63:32] | global_addr[120:64] | type=2
    v4u g0 = { 1u,
               lds,
               (unsigned)(ga & 0xFFFFFFFFull),
               (unsigned)((ga >> 32) & 0x01FFFFFFull) | 0x80000000u };

    // D# group 1: mask=0, data_size=2 (4B), dims/strides in float units
    unsigned td0 = (unsigned)tw;           // tensor_dim0 (row length, floats)
    unsigned td1 = 2u;                     // tensor_dim1 (2 source rows)
    unsigned tl0 = (unsigned)tw;           // tile_dim0
    unsigned tl1 = 2u;                     // tile_dim1
    unsigned st0 = (unsigned)(Ws * 3);     // tensor_dim0_stride (floats)
    v8i g1 = { (int)0x20000u,                                        // data_size=4B
               (int)((td0 & 0xFFFFu) << 16),                         // td0[15:0]
               (int)(((td0 >> 16) & 0xFFFFu) | ((td1 & 0xFFFFu) << 16)),
               (int)(((td1 >> 16) & 0xFFFFu) | (tl0 << 16)),         // tile_dim0
               (int)(tl1 & 0xFFFFu),                                 // tile_dim1 (tile_dim2=0)
               (int)st0,                                             // stride0[31:0]
               0,                                                    // stride0[47:32]|stride1[15:0]
               0 };                                                  // stride1[47:16]
    v4i z4 = {0, 0, 0, 0};
#if __has_include(<hip/amd_detail/amd_gfx1250_TDM.h>)
    v8i z8 = {0, 0, 0, 0, 0, 0, 0, 0};
    __builtin_amdgcn_tensor_load_to_lds(g0, g1, z4, z4, z8, 0);
#else
    __builtin_amdgcn_tensor_load_to_lds(g0, g1, z4, z4, 0);
#endif
    __builtin_amdgcn_s_wait_tensorcnt(0);
  }
  __syncthreads();

  if (tid < nout) {
    const int c0 = tid * 6;  // 2 source texels * 3 ch
    F3 r;
    r.x = 0.25f * (tile[c0 + 0] + tile[c0 + 3] + tile[tw + c0 + 0] + tile[tw + c0 + 3]);
    r.y = 0.25f * (tile[c0 + 1] + tile[c0 + 4] + tile[tw + c0 + 1] + tile[tw + c0 + 4]);
    r.z = 0.25f * (tile[c0 + 2] + tile[c0 + 5] + tile[tw + c0 + 2] + tile[tw + c0 + 5]);
    float* op = dst + ((size_t)yo * Wd + x0 + tid) * 3;
    op[0] = r.x; op[1] = r.y; op[2] = r.z;
  }
#else
  // --- Fallback: direct gathers ------------------------------------------
  if (tid < nout) {
    const int xo = x0 + tid;
    const float* r0 = src + ((size_t)(yo * 2) * Ws + (size_t)xo * 2) * 3;
    const float* r1 = r0 + (size_t)Ws * 3;
    F3 r;
    r.x = 0.25f * (r0[0] + r0[3] + r1[0] + r1[3]);
    r.y = 0.25f * (r0[1] + r0[4] + r1[1] + r1[4]);
    r.z = 0.25f * (r0[2] + r0[5] + r1[2] + r1[5]);
    float* op = dst + ((size_t)yo * Wd + xo) * 3;
    op[0] = r.x; op[1] = r.y; op[2] = r.z;
  }
#endif
}

// ---------------------------------------------------------------------------
// Phase 2: LOD + bilinear-wrap + trilinear lerp
// ---------------------------------------------------------------------------
__device__ __forceinline__ F3 fetch3(const float* __restrict__ base, int D,
                                     int iy, int ix) {
  const float* p = base + ((size_t)iy * D + ix) * 3;
  F3 r; r.x = p[0]; r.y = p[1]; r.z = p[2]; return r;   // global_load_b96
}

__device__ __forceinline__ F3 bilin_wrap(const float* __restrict__ base, int l,
                                         float u, float v) {
  const int   D = TEX_DIM >> l;
  const int   m = D - 1;                // power-of-two wrap mask
  const float x = u * (float)D - 0.5f;
  const float y = v * (float)D - 0.5f;
  const float xf = floorf(x), yf = floorf(y);
  const float fx = x - xf,    fy = y - yf;
  const int ix0 = ((int)xf) & m, iy0 = ((int)yf) & m;
  const int ix1 = (ix0 + 1) & m, iy1 = (iy0 + 1) & m;
  F3 v00 = fetch3(base, D, iy0, ix0);
  F3 v01 = fetch3(base, D, iy0, ix1);
  F3 v10 = fetch3(base, D, iy1, ix0);
  F3 v11 = fetch3(base, D, iy1, ix1);
  F3 r;
  r.x = (v00.x * (1.f - fx) + v01.x * fx) * (1.f - fy) + (v10.x * (1.f - fx) + v11.x * fx) * fy;
  r.y = (v00.y * (1.f - fx) + v01.y * fx) * (1.f - fy) + (v10.y * (1.f - fx) + v11.y * fx) * fy;
  r.z = (v00.z * (1.f - fx) + v01.z * fx) * (1.f - fy) + (v10.z * (1.f - fx) + v11.z * fx) * fy;
  return r;
}

__global__ void tex_sample_kernel(const float* __restrict__ tex,
                                  const float* __restrict__ ws,
                                  const float* __restrict__ uv,
                                  const float* __restrict__ uv_da,
                                  float* __restrict__ out,
                                  MipOff offs) {
  const int idx = blockIdx.x * blockDim.x + threadIdx.x;
  if (idx >= OUT_PIX) return;

  const float u = uv[(size_t)idx * 2 + 0];
  const float v = uv[(size_t)idx * 2 + 1];
  const float dsdx = uv_da[(size_t)idx * 4 + 0] * (float)TEX_DIM;
  const float dtdx = uv_da[(size_t)idx * 4 + 1] * (float)TEX_DIM;
  const float dsdy = uv_da[(size_t)idx * 4 + 2] * (float)TEX_DIM;
  const float dtdy = uv_da[(size_t)idx * 4 + 3] * (float)TEX_DIM;

  const float d2 = fmaxf(dsdx * dsdx + dtdx * dtdx, dsdy * dsdy + dtdy * dtdy);
  float level = 0.5f * log2f(fmaxf(d2, 1e-20f));
  level = fminf(fmaxf(level, 0.0f), (float)(N_LEVELS - 1));
  const int   l0 = (int)floorf(level);
  const int   l1 = min(l0 + 1, N_LEVELS - 1);
  const float f  = level - (float)l0;

  const float* b0 = (l0 == 0) ? tex : (ws + offs.off[l0]);
  const float* b1 = (l1 == 0) ? tex : (ws + offs.off[l1]);
  F3 s0 = bilin_wrap(b0, l0, u, v);
  F3 s1 = bilin_wrap(b1, l1, u, v);

  F3 r;
  r.x = s0.x * (1.f - f) + s1.x * f;
  r.y = s0.y * (1.f - f) + s1.y * f;
  r.z = s0.z * (1.f - f) + s1.z * f;

  float* op = out + (size_t)idx * 3;
  op[0] = r.x; op[1] = r.y; op[2] = r.z;
}

// ---------------------------------------------------------------------------
// Host launcher
// ---------------------------------------------------------------------------
extern "C" void kernel_launch(void* const* d_in, const int* in_sizes, int n_in,
                              void* d_out, int out_size, void* d_ws, size_t ws_size,
                              hipStream_t stream) {
  const float* tex   = (const float*)d_in[0];   // [1,2048,2048,3] f32
  const float* uv    = (const float*)d_in[1];   // [1,1024,1024,2] f32
  const float* uv_da = (const float*)d_in[2];   // [1,1024,1024,4] f32
  float*       out   = (float*)d_out;           // [1,1024,1024,3] f32
  float*       ws    = (float*)d_ws;            // mip pyramid, ~16.8 MB

  // Float offsets of levels 1..11 inside ws.
  MipOff offs;
  unsigned acc = 0;
  offs.off[0] = 0;
  for (int l = 1; l < N_LEVELS; ++l) {
    offs.off[l] = acc;
    const unsigned D = (unsigned)(TEX_DIM >> l);
    acc += D * D * 3u;
  }

  // Phase 1: build the pyramid (11 dependent launches on `stream`).
  const float* src = tex;
  for (int l = 1; l < N_LEVELS; ++l) {
    const int Ws_ = TEX_DIM >> (l - 1);
    const int Wd  = TEX_DIM >> l;
    float* dst = ws + offs.off[l];
    dim3 grid((Wd + POOL_BW - 1) / POOL_BW, Wd);
    mip_pool_kernel<<<grid, POOL_BW, 0, stream>>>(src, dst, Ws_, Wd);
    src = dst;
  }

  // Phase 2: sample (1M pixels, 8 waves/block).
  const int threads = 256;
  const int blocks  = (OUT_PIX + threads - 1) / threads;
  tex_sample_kernel<<<blocks, threads, 0, stream>>>(tex, ws, uv, uv_da, out, offs);
}